// Upsample_MLP_16836271800447
// MI455X (gfx1250) — compile-verified
//
#include <hip/hip_runtime.h>

// ---------------- types ----------------
typedef float v2f  __attribute__((ext_vector_type(2)));
typedef float v8f  __attribute__((ext_vector_type(8)));
typedef unsigned int u32x4 __attribute__((ext_vector_type(4)));
typedef int i32x4 __attribute__((ext_vector_type(4)));
typedef int i32x8 __attribute__((ext_vector_type(8)));

// ---------------- problem constants ----------------
#define Bn   4
#define Hn   256
#define Wn   256
#define HWn  65536
#define OUTC 64
#define HIDC 32

// ---------------- workspace layout (float offsets) ----------------
// All B-matrix weight blocks are stored K-pair interleaved: element (k,n) of a
// [K][N] matrix lives at ((k/2)*N + n)*2 + (k&1), so the (k,k+1) pair a lane
// needs for one f32-WMMA step is one aligned 8-byte LDS load.
#define WS_K1W   0        // 2 * 20480 : per row-parity i: [Wf_j0 8192][Wf_j1 8192][yw 4096]
#define WS_K3W   40960    // 5216      : [w1p 2048][w2 1024][w3 2048][b2 32][b3 64]
#define WS_EVC1  46208    // 256  : per-batch mlp1 ev contribution [4][64]
#define WS_BIASF 46464    // 64   : fused upconv_b@mlp1_w + mlp1_b
#define WS_EVC3  46528    // 128  : per-batch mlp3 layer1 ev + b1 [4][32]
#define WS_STATS 46656    // 8    : [b][{sum,sumsq}]
#define WS_MUSIG 46664    // 8    : [b][{mu, inv_std}]

// K1 LDS (floats)
#define SM1_YW   16384
#define SM1_BIAS 20480
#define SM1_RED  20544
#define SM1_TOT  20552
// K3 LDS (floats)
#define W1O 0
#define W2O 2048
#define W3O 3072
#define B2O 5120
#define B3O 5152
#define EVC3S 5248
#define TWO 5280           // per-wave 16x(32+pad2) transpose tiles, stride 34
#define SM3_TOT (5280 + 8*544)

// ---------------- CDNA5 f32 WMMA ----------------
static __device__ __forceinline__ v8f wmma4(v2f a, v2f b, v8f c) {
  // D(16x16,f32) = A(16x4,f32) * B(4x16,f32) + C
  return __builtin_amdgcn_wmma_f32_16x16x4_f32(false, a, false, b, (short)0, c,
                                               false, false);
}

static __device__ __forceinline__ v2f ldsv2(const float* p) {
  return *(const v2f*)p;   // 8B-aligned by construction -> ds_load_b64
}

// ---------------- TDM: DMA a flat run of f32 from global into LDS ----------------
#if __has_builtin(__builtin_amdgcn_tensor_load_to_lds)
#define HAVE_TDM 1
static __device__ __forceinline__ void tdm_load_1d(const float* g, unsigned lds_byte_off,
                                                   unsigned nelem) {
  unsigned long long ga = (unsigned long long)(const void*)g;
  u32x4 g0 = { 1u,                                  // count=1 (valid user descriptor)
               lds_byte_off,                        // lds_addr
               (unsigned)ga,                        // global_addr[31:0]
               (unsigned)(((ga >> 32) & 0x1FFFFFFull) | (2u << 30)) }; // addr[56:32]|type=2
  i32x8 g1 = { (int)(2u << 16),                     // data_size = 4 bytes
               (int)((nelem & 0xFFFFu) << 16),      // tensor_dim0[15:0]
               (int)(((nelem >> 16) & 0xFFFFu) | (1u << 16)), // dim0 hi | tensor_dim1=1
               (int)((nelem & 0xFFFFu) << 16),      // tile_dim0 (1-D tile)
               0,                                   // tile_dim1/2 unused
               (int)nelem,                          // tensor_dim0_stride
               0, 0 };
  i32x4 z4 = { 0, 0, 0, 0 };
  i32x8 z8 = { 0, 0, 0, 0, 0, 0, 0, 0 };
  __builtin_amdgcn_tensor_load_to_lds(g0, g1, z4, z4, z8, 0);
}
#endif

// preload a contiguous f32 run into LDS; wave0 drives the TDM, others skip (scalar branch)
static __device__ __forceinline__ void preload_lds(float* dst, const float* src, int n,
                                                   int tid) {
#if HAVE_TDM
  if (__builtin_amdgcn_readfirstlane((int)tid) == 0) {
    tdm_load_1d(src, (unsigned)(size_t)(void*)dst, (unsigned)n);
    __builtin_amdgcn_s_wait_tensorcnt(0);
  }
#else
  for (int o = tid; o < n; o += 256) dst[o] = src[o];
#endif
}

// pair-interleaved index for a [K][N] B-block
static __device__ __forceinline__ int pidx(int k, int n, int N) {
  return ((k >> 1) * N + n) * 2 + (k & 1);
}

// =====================================================================
// prep: fuse ConvT(2x2,s2) into mlp1, fold ev scalars/biases, pack, zero stats
// =====================================================================
__global__ __launch_bounds__(256) void prep_kernel(
    const float* __restrict__ sv,  const float* __restrict__ bv,
    const float* __restrict__ upw, const float* __restrict__ upb,
    const float* __restrict__ m1w, const float* __restrict__ m1b,
    const float* __restrict__ w1,  const float* __restrict__ b1,
    const float* __restrict__ w2,  const float* __restrict__ b2,
    const float* __restrict__ w3,  const float* __restrict__ b3,
    float* __restrict__ ws) {
  const int tid = threadIdx.x;
  // Wf[i][j][k(128)][n(64)] = sum_c upw[k][c][i][j] * m1w[c][n]  (pair-interleaved)
  for (int o = tid; o < 32768; o += 256) {
    int ij = o >> 13, rem = o & 8191;
    int k = rem >> 6, n = rem & 63;
    int pi = ij >> 1, pj = ij & 1;
    float acc = 0.f;
#pragma unroll 4
    for (int c = 0; c < 64; ++c)
      acc += upw[((k * 64 + c) * 2 + pi) * 2 + pj] * m1w[c * 64 + n];
    ws[WS_K1W + pi * 20480 + pj * 8192 + pidx(k, n, 64)] = acc;
  }
  // y-path weights (mlp1 rows 64..127), duplicated after each parity block
  for (int o = tid; o < 4096; o += 256) {
    int k = o >> 6, n = o & 63;
    float v = m1w[(64 + k) * 64 + n];
    ws[WS_K1W + SM1_YW + pidx(k, n, 64)] = v;
    ws[WS_K1W + 20480 + SM1_YW + pidx(k, n, 64)] = v;
  }
  // mlp3 weight pack (pair-interleaved)
  for (int o = tid; o < 2048; o += 256) {            // w1 rows 0..63, [64][32]
    int k = o >> 5, n = o & 31;
    ws[WS_K3W + W1O + pidx(k, n, 32)] = w1[o];
  }
  for (int o = tid; o < 1024; o += 256) {            // w2 [32][32]
    int k = o >> 5, n = o & 31;
    ws[WS_K3W + W2O + pidx(k, n, 32)] = w2[o];
  }
  for (int o = tid; o < 2048; o += 256) {            // w3 [32][64]
    int k = o >> 6, n = o & 63;
    ws[WS_K3W + W3O + pidx(k, n, 64)] = w3[o];
  }
  if (tid < 32) ws[WS_K3W + B2O + tid] = b2[tid];
  if (tid < 64) ws[WS_K3W + B3O + tid] = b3[tid];
  // ev contributions
  if (tid < 256) {
    int b_ = tid >> 6, n = tid & 63;
    ws[WS_EVC1 + tid] = sv[b_] * m1w[128 * 64 + n] + bv[b_] * m1w[129 * 64 + n];
  }
  if (tid < 64) {
    float acc = m1b[tid];
#pragma unroll 4
    for (int c = 0; c < 64; ++c) acc += upb[c] * m1w[c * 64 + tid];
    ws[WS_BIASF + tid] = acc;
  }
  if (tid < 128) {
    int b_ = tid >> 5, n = tid & 31;
    ws[WS_EVC3 + tid] = sv[b_] * w1[64 * 32 + n] + bv[b_] * w1[65 * 32 + n] + b1[n];
  }
  if (tid < 8) ws[WS_STATS + tid] = 0.f;   // re-zeroed every launch (graph-safe)
}

// =====================================================================
// K1: fused upconv+concat+mlp1+ReLU -> h1 into d_out; LN sum/sumsq atomics
// grid: B*H blocks of 256 (8 waves); wave = 16 same-parity pixels of row h
// =====================================================================
__global__ __launch_bounds__(256) void k1_kernel(
    const float* __restrict__ x, const float* __restrict__ y,
    float* __restrict__ ws, float* __restrict__ out) {
  extern __shared__ float smem[];
  const int tid = threadIdx.x;
  const int bb = blockIdx.x >> 8;
  const int h  = blockIdx.x & 255;
  const int i  = h & 1;

  preload_lds(smem, ws + WS_K1W + (size_t)i * 20480, 20480, tid);
  if (tid < 64) smem[SM1_BIAS + tid] = ws[WS_BIASF + tid] + ws[WS_EVC1 + bb * 64 + tid];
  if (tid < 2) smem[SM1_RED + tid] = 0.f;
  __syncthreads();

  const int lane = tid & 31, wv = tid >> 5;
  const int ml = lane & 15, hi = lane >> 4;
  float lsum = 0.f, lsq = 0.f;

  for (int it = 0; it < 2; ++it) {
    const int j = it;                 // output-column parity
    const int mbase = wv * 16;
    const int m = mbase + ml;         // x column / compacted pixel index
    const float* xb = x + (size_t)bb * 2097152 + (size_t)(h >> 1) * 128 + m; // +k*16384
    const float* yb = y + (size_t)bb * 4194304 + (size_t)h * 256 + (2 * m + j); // +k*65536

    v8f z8 = {0, 0, 0, 0, 0, 0, 0, 0};
    v8f acc[4] = {z8, z8, z8, z8};

    // x path: K = 128 fused weights Wf[i][j] (pair-interleaved B)
    const float* wf = smem + j * 8192;
#pragma unroll 2
    for (int ks = 0; ks < 32; ++ks) {
      const int k0 = ks * 4 + hi * 2;       // k0/2 = 2*ks + hi
      v2f a;
      a.x = xb[(size_t)k0 * 16384];
      a.y = xb[(size_t)(k0 + 1) * 16384];
      const float* wrow = wf + ((2 * ks + hi) * 64 + ml) * 2;
#pragma unroll
      for (int t = 0; t < 4; ++t)
        acc[t] = wmma4(a, ldsv2(wrow + t * 32), acc[t]);
    }
    // y path: K = 64
    const float* yw = smem + SM1_YW;
#pragma unroll 2
    for (int ks = 0; ks < 16; ++ks) {
      const int k0 = ks * 4 + hi * 2;
      v2f a;
      a.x = yb[(size_t)k0 * 65536];
      a.y = yb[(size_t)(k0 + 1) * 65536];
      const float* wrow = yw + ((2 * ks + hi) * 64 + ml) * 2;
#pragma unroll
      for (int t = 0; t < 4; ++t)
        acc[t] = wmma4(a, ldsv2(wrow + t * 32), acc[t]);
    }
    // epilogue: bias+ev, ReLU, store h1 (NCHW, parity-interleaved), LN partials
#pragma unroll
    for (int t = 0; t < 4; ++t) {
      const int n = t * 16 + ml;
      const float bsv = smem[SM1_BIAS + n];
      float* op = out + ((size_t)bb * 64 + n) * 65536 + (size_t)h * 256;
#pragma unroll
      for (int r = 0; r < 8; ++r) {
        float v = fmaxf(acc[t][r] + bsv, 0.f);
        lsum += v;
        lsq += v * v;
        op[2 * (mbase + hi * 8 + r) + j] = v;
      }
    }
  }

  // wave32 reduce -> LDS -> global f32 atomics
#pragma unroll
  for (int o = 16; o; o >>= 1) {
    lsum += __shfl_xor(lsum, o);
    lsq += __shfl_xor(lsq, o);
  }
  if (lane == 0) {
    atomicAdd(&smem[SM1_RED + 0], lsum);
    atomicAdd(&smem[SM1_RED + 1], lsq);
  }
  __syncthreads();
  if (tid < 2) unsafeAtomicAdd(&ws[WS_STATS + bb * 2 + tid], smem[SM1_RED + tid]);
}

// =====================================================================
// K2: mean / inv_std per batch sample
// =====================================================================
__global__ void stats_kernel(float* __restrict__ ws) {
  const int t = threadIdx.x;
  if (t < 4) {
    const float N = 4194304.f;  // 64*256*256
    float sum = ws[WS_STATS + t * 2], sq = ws[WS_STATS + t * 2 + 1];
    float mu = sum / N;
    float var = fmaxf(sq / N - mu * mu, 0.f);
    ws[WS_MUSIG + t * 2] = mu;
    ws[WS_MUSIG + t * 2 + 1] = rsqrtf(var + 1e-5f);
  }
}

// =====================================================================
// K3: LayerNorm affine + mlp3 chain + ReLU + residual, in-place on d_out
// grid: B*H blocks of 256; wave = 16 consecutive pixels
// =====================================================================
__global__ __launch_bounds__(256) void k3_kernel(
    const float* __restrict__ lnsc, const float* __restrict__ lnbs,
    const float* __restrict__ ws, float* __restrict__ out) {
  extern __shared__ float smem[];
  const int tid = threadIdx.x;
  const int bb = blockIdx.x >> 8;
  const int h  = blockIdx.x & 255;

  preload_lds(smem, ws + WS_K3W, 5216, tid);
  if (tid < 32) smem[EVC3S + tid] = ws[WS_EVC3 + bb * 32 + tid];
  __syncthreads();

  const float mu = ws[WS_MUSIG + bb * 2];
  const float istd = ws[WS_MUSIG + bb * 2 + 1];
  const int lane = tid & 31, wv = tid >> 5;
  const int ml = lane & 15, hi = lane >> 4;
  float* tw = smem + TWO + wv * 544;   // per-wave 16x34 transpose tile

  for (int it = 0; it < 2; ++it) {
    const int pix0 = (wv + it * 8) * 16;
    const int p = pix0 + ml;
    const float* hb = out + (size_t)bb * 4194304 + (size_t)h * 256 + p;   // +k*65536
    const float* sb = lnsc + (size_t)h * 256 + p;
    const float* qb = lnbs + (size_t)h * 256 + p;

    v8f z8 = {0, 0, 0, 0, 0, 0, 0, 0};
    // ---- layer1: normed(64) @ w1p -> 32
    v8f acc1[2] = {z8, z8};
#pragma unroll 2
    for (int ks = 0; ks < 16; ++ks) {
      const int k0 = ks * 4 + hi * 2;
      const size_t o0 = (size_t)k0 * 65536, o1 = o0 + 65536;
      v2f a;
      a.x = (hb[o0] - mu) * istd * sb[o0] + qb[o0];
      a.y = (hb[o1] - mu) * istd * sb[o1] + qb[o1];
      const float* wrow = smem + W1O + ((2 * ks + hi) * 32 + ml) * 2;
#pragma unroll
      for (int t = 0; t < 2; ++t)
        acc1[t] = wmma4(a, ldsv2(wrow + t * 32), acc1[t]);
    }
    // D->A transpose via LDS (+ev/b1)
#pragma unroll
    for (int t = 0; t < 2; ++t) {
      const int n = t * 16 + ml;
      const float e = smem[EVC3S + n];
#pragma unroll
      for (int r = 0; r < 8; ++r) tw[(r + 8 * hi) * 34 + n] = acc1[t][r] + e;
    }
    __builtin_amdgcn_wave_barrier();

    // ---- layer2: 32 @ w2 -> 32
    v8f acc2[2] = {z8, z8};
#pragma unroll 2
    for (int ks = 0; ks < 8; ++ks) {
      const int k0 = ks * 4 + hi * 2;
      v2f a = ldsv2(tw + ml * 34 + k0);
      const float* wrow = smem + W2O + ((2 * ks + hi) * 32 + ml) * 2;
#pragma unroll
      for (int t = 0; t < 2; ++t)
        acc2[t] = wmma4(a, ldsv2(wrow + t * 32), acc2[t]);
    }
    __builtin_amdgcn_wave_barrier();
#pragma unroll
    for (int t = 0; t < 2; ++t) {
      const int n = t * 16 + ml;
      const float e = smem[B2O + n];
#pragma unroll
      for (int r = 0; r < 8; ++r) tw[(r + 8 * hi) * 34 + n] = acc2[t][r] + e;
    }
    __builtin_amdgcn_wave_barrier();

    // ---- layer3: 32 @ w3 -> 64, ReLU, + identity, store NCHW
    v8f acc3[4] = {z8, z8, z8, z8};
#pragma unroll 2
    for (int ks = 0; ks < 8; ++ks) {
      const int k0 = ks * 4 + hi * 2;
      v2f a = ldsv2(tw + ml * 34 + k0);
      const float* wrow = smem + W3O + ((2 * ks + hi) * 64 + ml) * 2;
#pragma unroll
      for (int t = 0; t < 4; ++t)
        acc3[t] = wmma4(a, ldsv2(wrow + t * 32), acc3[t]);
    }
#pragma unroll
    for (int t = 0; t < 4; ++t) {
      const int n = t * 16 + ml;
      const float b3v = smem[B3O + n];
      float* ob = out + ((size_t)bb * 64 + n) * 65536 + (size_t)h * 256 + pix0 + 8 * hi;
      float4 i0 = *(const float4*)ob;
      float4 i1 = *(const float4*)(ob + 4);
      float4 r0, r1;
      r0.x = fmaxf(acc3[t][0] + b3v, 0.f) + i0.x;
      r0.y = fmaxf(acc3[t][1] + b3v, 0.f) + i0.y;
      r0.z = fmaxf(acc3[t][2] + b3v, 0.f) + i0.z;
      r0.w = fmaxf(acc3[t][3] + b3v, 0.f) + i0.w;
      r1.x = fmaxf(acc3[t][4] + b3v, 0.f) + i1.x;
      r1.y = fmaxf(acc3[t][5] + b3v, 0.f) + i1.y;
      r1.z = fmaxf(acc3[t][6] + b3v, 0.f) + i1.z;
      r1.w = fmaxf(acc3[t][7] + b3v, 0.f) + i1.w;
      *(float4*)ob = r0;
      *(float4*)(ob + 4) = r1;
    }
  }
}

// =====================================================================
extern "C" void kernel_launch(void* const* d_in, const int* in_sizes, int n_in,
                              void* d_out, int out_size, void* d_ws, size_t ws_size,
                              hipStream_t stream) {
  const float* x   = (const float*)d_in[0];
  const float* y   = (const float*)d_in[1];
  const float* sv  = (const float*)d_in[2];
  const float* bv  = (const float*)d_in[3];
  const float* upw = (const float*)d_in[4];
  const float* upb = (const float*)d_in[5];
  const float* m1w = (const float*)d_in[6];
  const float* m1b = (const float*)d_in[7];
  const float* w1  = (const float*)d_in[8];
  const float* b1  = (const float*)d_in[9];
  const float* w2  = (const float*)d_in[10];
  const float* b2  = (const float*)d_in[11];
  const float* w3  = (const float*)d_in[12];
  const float* b3  = (const float*)d_in[13];
  const float* lnsc = (const float*)d_in[14];
  const float* lnbs = (const float*)d_in[15];
  float* ws  = (float*)d_ws;
  float* out = (float*)d_out;

  prep_kernel<<<1, 256, 0, stream>>>(sv, bv, upw, upb, m1w, m1b, w1, b1, w2, b2, w3,
                                     b3, ws);
  k1_kernel<<<Bn * Hn, 256, SM1_TOT * sizeof(float), stream>>>(x, y, ws, out);
  stats_kernel<<<1, 64, 0, stream>>>(ws);
  k3_kernel<<<Bn * Hn, 256, SM3_TOT * sizeof(float), stream>>>(lnsc, lnbs, ws, out);
}